// PhraseSim_43499428773890
// MI455X (gfx1250) — compile-verified
//
#include <hip/hip_runtime.h>
#include <hip/hip_bf16.h>
#include <math.h>

// ---------------- problem constants (from reference) ----------------
#define L1N   1024
#define L2N   1024
#define BN    32
#define HN    128
#define KTOP  128
#define NSL   4
#define DN    (KTOP * NSL)          // 512
#define M1    (L1N * BN)            // 32768 rows of H1 viewed as dense [M1, H]
#define CANDCAP 2048
#define BT_STRIDE 132               // padded LDS row stride (floats): banks = 4*j + kb

typedef __attribute__((ext_vector_type(2))) float v2f;
typedef __attribute__((ext_vector_type(8))) float v8f;

// ---------------- workspace layout (bytes) ----------------
#define WS_AW        0ull                                   // NS*M1*H f32 = 67,108,864
#define WS_HIST      (WS_AW + (size_t)NSL * M1 * HN * 4)    // B*NS*1024 u32 = 524,288
#define WS_META      (WS_HIST + (size_t)BN * NSL * 1024 * 4)// B*NS*4 i32 = 2,048
#define WS_CNT       (WS_META + (size_t)BN * NSL * 4 * 4)   // B*NS*2 u32 = 1,024
#define WS_HIBUF     (WS_CNT + (size_t)BN * NSL * 2 * 4)    // B*NS*128 f32 = 65,536
#define WS_CANDBUF   (WS_HIBUF + (size_t)BN * NSL * KTOP * 4)   // B*NS*2048 f32
#define WS_RES       (WS_CANDBUF + (size_t)BN * NSL * CANDCAP * 4) // B*D f32
#define WS_ZERO_BYTES ((size_t)BN * NSL * 1024 * 4 + (size_t)BN * NSL * 4 * 4 + (size_t)BN * NSL * 2 * 4)

// Async global->LDS copy of 16B per lane (CDNA5 ASYNCcnt path).
__device__ __forceinline__ void async_copy_b128(unsigned lds_byte_addr, unsigned long long gaddr) {
  asm volatile("global_load_async_to_lds_b128 %0, %1, off"
               :: "v"(lds_byte_addr), "v"(gaddr) : "memory");
}
__device__ __forceinline__ void wait_asynccnt0() {
  asm volatile("s_wait_asynccnt 0x0" ::: "memory");
}

// =====================================================================
// Kernel 1: AW[s] = outputs1(as [M1,H]) @ Ws[s]   (fp32 WMMA 16x16x4)
// grid: NS * (M1/128) blocks, 256 threads (8 waves); wave -> 16x128 strip
// Ws is 64KB/slice and hot in WGP$/L2; direct loads are fine here.
// =====================================================================
__global__ void k_h1w(const float* __restrict__ X, const float* __restrict__ Ws,
                      float* __restrict__ AW) {
  const int lane = threadIdx.x & 31;
  const int wave = threadIdx.x >> 5;
  const int l15  = lane & 15;
  const int hihf = lane >> 4;                 // 0 or 1 (half-wave)
  const int bps  = M1 / 128;                  // 256 blocks per slice
  const int s    = blockIdx.x / bps;
  const int mblk = blockIdx.x % bps;
  const int m0   = mblk * 128 + wave * 16;

  const float* W    = Ws + (size_t)s * HN * HN;
  const float* Arow = X + (size_t)(m0 + l15) * HN;    // A fragment row for this lane
  float* Obase      = AW + (size_t)s * M1 * HN;

  v8f acc[8] = {};
  for (int kk = 0; kk < HN / 4; ++kk) {
    const int kb = kk * 4 + hihf * 2;
    v2f a = *(const v2f*)(Arow + kb);          // A: M=l15, K=kb..kb+1
#pragma unroll
    for (int n = 0; n < 8; ++n) {
      v2f bf;
      bf.x = W[(size_t)kb * HN + n * 16 + l15];        // B: K=kb,   N=n*16+l15
      bf.y = W[(size_t)(kb + 1) * HN + n * 16 + l15];  // B: K=kb+1, N=n*16+l15
      acc[n] = __builtin_amdgcn_wmma_f32_16x16x4_f32(
          false, a, false, bf, (short)0, acc[n], false, false);
    }
  }
#pragma unroll
  for (int n = 0; n < 8; ++n) {
#pragma unroll
    for (int r = 0; r < 8; ++r) {
      const int row = m0 + r + 8 * hihf;       // C/D layout: M = r + 8*half
      Obase[(size_t)row * HN + n * 16 + l15] = acc[n][r];
    }
  }
}

// =====================================================================
// Kernel 2: score tiles S = AW[s,b] @ H2[b]^T + bias, relu, mask.
// B-tile (H2 rows) is shared by all 8 waves -> stage it in LDS via
// async global->LDS B128 copies (two 64-row halves, 33KB), then feed
// WMMA B-fragments from LDS (conflict-free ds_load_b64, stride 132).
// MODE 0: 1024-bin histogram of fp32 bit-key (bits>>22).
// MODE 1: collect values with bin>T (top set) and bin==T (candidates).
// grid: (B*NS) * 64 tile-blocks, 256 threads; wave -> 16 rows x 128 cols
// =====================================================================
template <int MODE>
__global__ void k_scores(const float* __restrict__ AW, const float* __restrict__ H2,
                         const int* __restrict__ mask1, const int* __restrict__ mask2,
                         const float* __restrict__ bsv,
                         unsigned* __restrict__ hist,
                         const int* __restrict__ meta,
                         unsigned* __restrict__ counters,
                         float* __restrict__ hi_buf,
                         float* __restrict__ cand_buf) {
  __shared__ float Bt[64 * BT_STRIDE];        // 33,792 B staged B half-tile
  __shared__ unsigned lhist[1024];            // 4 KB
  const int lane = threadIdx.x & 31;
  const int wave = threadIdx.x >> 5;
  const int l15  = lane & 15;
  const int hihf = lane >> 4;

  const int tile = blockIdx.x & 63;
  const int bs   = blockIdx.x >> 6;           // bs = b*NS + s
  const int b    = bs >> 2;
  const int s    = bs & 3;
  const int ti   = tile >> 3, tj = tile & 7;

  if (MODE == 0) {
    for (int i = threadIdx.x; i < 1024; i += blockDim.x) lhist[i] = 0u;
    __syncthreads();
  }

  const float* Abase = AW + (size_t)s * M1 * HN + (size_t)b * HN;  // (i,h)@ i*B*H + h
  const float* Bbase = H2 + (size_t)b * HN;                        // (j,h)@ j*B*H + h
  const int i0 = ti * 128 + wave * 16;
  const int j0 = tj * 128;
  const float bias = bsv[s];
  const float* Arow = Abase + (size_t)(i0 + l15) * (BN * HN);
  const unsigned bt_base = (unsigned)(size_t)&Bt[0];   // LDS byte offset (low 32 bits)

  v8f acc[8] = {};
#pragma unroll
  for (int half = 0; half < 2; ++half) {
    // ---- stage 64 B-rows (j0 + half*64 ..) into LDS: 8 rows per wave,
    //      one async B128 instruction per row (32 lanes x 16B = 512B row)
#pragma unroll
    for (int r = 0; r < 8; ++r) {
      const int row_local = wave * 8 + r;
      const int j = j0 + half * 64 + row_local;
      const unsigned long long ga =
          (unsigned long long)(const void*)(Bbase + (size_t)j * (BN * HN)) +
          (unsigned)lane * 16u;
      const unsigned la = bt_base + (unsigned)(row_local * BT_STRIDE) * 4u +
                          (unsigned)lane * 16u;
      async_copy_b128(la, ga);
    }
    wait_asynccnt0();        // own copies landed in LDS
    __syncthreads();         // all waves' copies visible

    // ---- WMMA over K=128 for the 4 column-subtiles of this half
    for (int kk = 0; kk < HN / 4; ++kk) {
      const int kb = kk * 4 + hihf * 2;
      v2f a = *(const v2f*)(Arow + kb);
#pragma unroll
      for (int n = 0; n < 4; ++n) {
        const int jl = n * 16 + l15;                       // row within half-tile
        v2f bf = *(const v2f*)(&Bt[jl * BT_STRIDE + kb]);  // ds_load_b64, no conflicts
        acc[half * 4 + n] = __builtin_amdgcn_wmma_f32_16x16x4_f32(
            false, a, false, bf, (short)0, acc[half * 4 + n], false, false);
      }
    }
    __syncthreads();         // done reading Bt before next half overwrites it
  }

  int T = 0;
  if (MODE == 1) T = meta[bs * 4 + 0];

#pragma unroll
  for (int n = 0; n < 8; ++n) {
    const int j = j0 + n * 16 + l15;
    const bool vj = (mask2[(size_t)j * BN + b] == 0);
#pragma unroll
    for (int r = 0; r < 8; ++r) {
      if (!vj) continue;
      const int i = i0 + r + 8 * hihf;
      if (mask1[(size_t)i * BN + b] != 0) continue;
      float v = acc[n][r] + bias;
      v = v > 0.0f ? v : 0.0f;                          // relu (valid scores >= 0)
      const int bin = (int)(__float_as_uint(v) >> 22);  // monotone key, 1024 bins
      if (MODE == 0) {
        atomicAdd(&lhist[bin], 1u);
      } else {
        if (bin > T) {
          unsigned p = atomicAdd(&counters[bs * 2 + 0], 1u);
          if (p < (unsigned)KTOP) hi_buf[(size_t)bs * KTOP + p] = v;
        } else if (bin == T) {
          unsigned p = atomicAdd(&counters[bs * 2 + 1], 1u);
          if (p < (unsigned)CANDCAP) cand_buf[(size_t)bs * CANDCAP + p] = v;
        }
      }
    }
  }

  if (MODE == 0) {
    __syncthreads();
    for (int i = threadIdx.x; i < 1024; i += blockDim.x) {
      unsigned c = lhist[i];
      if (c) atomicAdd(&hist[(size_t)bs * 1024 + i], c);
    }
  }
}

// =====================================================================
// Kernel 2b: find threshold bin T per (b,s): largest T with
// count(bin > T) < K <= count(bin >= T). T=-1 if total valid < K.
// =====================================================================
__global__ void k_thresh(const unsigned* __restrict__ hist, int* __restrict__ meta) {
  const int bs = threadIdx.x;
  if (bs >= BN * NSL) return;
  const unsigned* h = hist + (size_t)bs * 1024;
  unsigned cum = 0;
  int T = -1, count_hi = 0;
  for (int bin = 1023; bin >= 0; --bin) {
    unsigned c = h[bin];
    if (cum + c >= (unsigned)KTOP) { T = bin; count_hi = (int)cum; break; }
    cum += c;
  }
  if (T < 0) count_hi = (int)cum;
  meta[bs * 4 + 0] = T;
  meta[bs * 4 + 1] = count_hi;
  meta[bs * 4 + 2] = (T >= 0) ? (KTOP - count_hi) : 0;
}

// ---------------- bitonic sort (descending), in LDS ----------------
__device__ void bitonic_desc(float* a, int n, int tid, int nt) {
  for (int k = 2; k <= n; k <<= 1) {
    for (int j = k >> 1; j > 0; j >>= 1) {
      __syncthreads();
      for (int i = tid; i < n; i += nt) {
        const int ixj = i ^ j;
        if (ixj > i) {
          const bool up = ((i & k) == 0);
          float x = a[i], y = a[ixj];
          if (up ? (x < y) : (x > y)) { a[i] = y; a[ixj] = x; }
        }
      }
    }
  }
  __syncthreads();
}

// =====================================================================
// Kernel 2d: per (b,s) sort hi-set and candidates, emit sorted top-K
// with last-finite fallback fill. grid: B*NS blocks, 256 threads.
// =====================================================================
__global__ void k_select(const int* __restrict__ meta, const unsigned* __restrict__ counters,
                         const float* __restrict__ hi_buf, const float* __restrict__ cand_buf,
                         float* __restrict__ res) {
  __shared__ float cand[CANDCAP];
  __shared__ float hiv[KTOP];
  const int bs = blockIdx.x;
  const int b = bs >> 2, s = bs & 3;
  const int tid = threadIdx.x;
  const int count_hi = meta[bs * 4 + 1];
  const int need     = meta[bs * 4 + 2];
  int ncand = (int)counters[bs * 2 + 1];
  if (ncand > CANDCAP) ncand = CANDCAP;

  for (int i = tid; i < CANDCAP; i += blockDim.x)
    cand[i] = (i < ncand) ? cand_buf[(size_t)bs * CANDCAP + i] : -INFINITY;
  for (int i = tid; i < KTOP; i += blockDim.x)
    hiv[i] = (i < count_hi) ? hi_buf[(size_t)bs * KTOP + i] : -INFINITY;
  __syncthreads();
  bitonic_desc(cand, CANDCAP, tid, blockDim.x);
  bitonic_desc(hiv, KTOP, tid, blockDim.x);

  const int take = need < ncand ? need : ncand;
  const int tot  = count_hi + take;
  float fillv = 0.0f;
  if (tot > 0) fillv = (tot - 1 < count_hi) ? hiv[tot - 1] : cand[tot - 1 - count_hi];

  float* out = res + (size_t)b * DN + (size_t)s * KTOP;
  for (int i = tid; i < KTOP; i += blockDim.x) {
    float v;
    if (i < count_hi)      v = hiv[i];
    else if (i < tot)      v = cand[i - count_hi];
    else                   v = fillv;
    out[i] = v;
  }
}

// =====================================================================
// Kernel 3: per-batch MLP: relu(res@W1+b1)@W2+b2 -> softmax.
// grid: B blocks, 256 threads. Tiny (16.8 MFLOP total).
// =====================================================================
__global__ void k_mlp(const float* __restrict__ res, const float* __restrict__ W1,
                      const float* __restrict__ b1, const float* __restrict__ W2,
                      const float* __restrict__ b2, float* __restrict__ out) {
  __shared__ float r[DN];
  __shared__ float h[DN];
  __shared__ float red0[256], red1[256];
  const int b = blockIdx.x;
  const int tid = threadIdx.x;
  for (int i = tid; i < DN; i += 256) r[i] = res[(size_t)b * DN + i];
  __syncthreads();
  for (int c = tid; c < DN; c += 256) {
    float acc = b1[c];
    for (int d = 0; d < DN; ++d) acc += r[d] * W1[(size_t)d * DN + c];
    h[c] = acc > 0.0f ? acc : 0.0f;
  }
  __syncthreads();
  float p0 = 0.f, p1 = 0.f;
  for (int c = tid; c < DN; c += 256) { p0 += h[c] * W2[c * 2]; p1 += h[c] * W2[c * 2 + 1]; }
  red0[tid] = p0; red1[tid] = p1;
  __syncthreads();
  for (int st = 128; st > 0; st >>= 1) {
    if (tid < st) { red0[tid] += red0[tid + st]; red1[tid] += red1[tid + st]; }
    __syncthreads();
  }
  if (tid == 0) {
    const float l0 = red0[0] + b2[0], l1 = red1[0] + b2[1];
    const float m = l0 > l1 ? l0 : l1;
    const float e0 = __expf(l0 - m), e1 = __expf(l1 - m);
    const float inv = 1.0f / (e0 + e1);
    out[b * 2 + 0] = e0 * inv;
    out[b * 2 + 1] = e1 * inv;
  }
}

// =====================================================================
extern "C" void kernel_launch(void* const* d_in, const int* in_sizes, int n_in,
                              void* d_out, int out_size, void* d_ws, size_t ws_size,
                              hipStream_t stream) {
  const float* X1  = (const float*)d_in[0];   // outputs1 [L1,B,H]
  const float* X2  = (const float*)d_in[1];   // outputs2 [L2,B,H]
  const int*   m1  = (const int*)d_in[2];     // mask1 [L1,B]
  const int*   m2  = (const int*)d_in[3];     // mask2 [L2,B]
  const float* Ws  = (const float*)d_in[4];   // [NS,H,H]
  const float* bsv = (const float*)d_in[5];   // [NS]
  const float* W1  = (const float*)d_in[6];   // [D,D]
  const float* b1  = (const float*)d_in[7];   // [D]
  const float* W2  = (const float*)d_in[8];   // [D,2]
  const float* b2  = (const float*)d_in[9];   // [2]

  char* ws = (char*)d_ws;
  float*    AW       = (float*)(ws + WS_AW);
  unsigned* hist     = (unsigned*)(ws + WS_HIST);
  int*      meta     = (int*)(ws + WS_META);
  unsigned* counters = (unsigned*)(ws + WS_CNT);
  float*    hi_buf   = (float*)(ws + WS_HIBUF);
  float*    cand_buf = (float*)(ws + WS_CANDBUF);
  float*    res      = (float*)(ws + WS_RES);

  // zero hist + meta + counters (contiguous) every call — deterministic
  hipMemsetAsync(hist, 0, WS_ZERO_BYTES, stream);

  // 1) H1W (fp32 WMMA)
  k_h1w<<<NSL * (M1 / 128), 256, 0, stream>>>(X1, Ws, AW);
  // 2) histogram pass (fp32 WMMA score tiles, async-LDS staged B)
  k_scores<0><<<BN * NSL * 64, 256, 0, stream>>>(AW, X2, m1, m2, bsv,
                                                 hist, meta, counters, hi_buf, cand_buf);
  // 3) threshold bin per (b,s)
  k_thresh<<<1, 128, 0, stream>>>(hist, meta);
  // 4) collect pass (fp32 WMMA score tiles, recomputed)
  k_scores<1><<<BN * NSL * 64, 256, 0, stream>>>(AW, X2, m1, m2, bsv,
                                                 hist, meta, counters, hi_buf, cand_buf);
  // 5) sort + select + fill
  k_select<<<BN * NSL, 256, 0, stream>>>(meta, counters, hi_buf, cand_buf, res);
  // 6) MLP head + softmax
  k_mlp<<<BN, 256, 0, stream>>>(res, W1, b1, W2, b2, (float*)d_out);
}